// SPDCNN_41059887350428
// MI455X (gfx1250) — compile-verified
//
#include <hip/hip_runtime.h>
#include <math.h>

// ---------------------------------------------------------------------------
// SPD-CNN pipeline for MI455X (gfx1250, wave32, WMMA + TDM + async->LDS).
//  k_wker    : Wker = V^T V + eps I, stored transposed as [K=200][R=16]
//  k_conv    : implicit-GEMM VALID conv via v_wmma_f32_16x16x4_f32; B matrix
//              staged to LDS with a single TENSOR_LOAD_TO_LDS DMA (TENSORcnt)
//  k_norm    : per-channel Frobenius norm (deterministic tree reduction)
//  k_matfunc : per-(b,r) LDS-resident matrix log (inverse scaling&squaring,
//              Newton-Schulz sqrts) -> 2x2 avg pool -> matrix exp (Taylor +
//              squaring), all GEMMs on WMMA f32; input tile staged via
//              global_load_async_to_lds_b128 (ASYNCcnt) so the transposed
//              (symmetrize) reads hit LDS, not HBM
//  k_cls     : 512x7440 x 7440x10 classifier with global_prefetch
// ---------------------------------------------------------------------------

typedef float    v2f __attribute__((ext_vector_type(2)));
typedef float    v8f __attribute__((ext_vector_type(8)));
typedef unsigned v4u __attribute__((ext_vector_type(4)));
typedef unsigned v8u __attribute__((ext_vector_type(8)));

#define EPS      1e-6f
#define LDA      65     // 64x64 LDS matrices, padded stride (bank spread)
#define LDB      33     // 32x32 LDS matrices
#define NS_ITERS 9      // Newton-Schulz iterations per matrix sqrt
#define N_SQRT   5      // inverse scaling-and-squaring depth for log
#define EXP_T    5      // squarings for exp (scale = 1/32)

// Low 32 bits of a generic pointer to an LDS object = LDS byte offset
// (ISA 10.2: LDS aperture check uses addr[63:32]).
__device__ __forceinline__ unsigned lds_offset(const void* p)
{
    return (unsigned)(unsigned long long)p;
}

// ---- Tensor Data Mover: 1-row tile of nelem f32, global -> LDS ------------
// D# per ISA 8.3/8.4: group0 = {count=1, lds_addr, global_addr, type=2},
// group1 = {data_size=4B, tensor_dim0=tile_dim0=nelem, tensor_dim1=tile_dim1=1,
// dim0_stride=nelem}. 2-D tensor -> descriptor groups 2/3 are NULL (omitted).
// Tracked on TENSORcnt; EXEC is ignored so issue from a single wave.
__device__ __forceinline__ void tdm_load_1d(unsigned lds_addr,
                                            const void* gptr,
                                            unsigned nelem)
{
    const unsigned long long ga = (unsigned long long)gptr;
    v4u g0;
    g0[0] = 1u;                                       // count=1 (user D#)
    g0[1] = lds_addr;                                 // lds_addr[31:0]
    g0[2] = (unsigned)ga;                             // global_addr[31:0]
    g0[3] = (unsigned)(ga >> 32) | 0x80000000u;       // ga[56:32] | type=2
    v8u g1;
    g1[0] = 2u << 16;                                 // data_size = 2 (4 bytes)
    g1[1] = (nelem & 0xFFFFu) << 16;                  // tensor_dim0[15:0]@48
    g1[2] = ((nelem >> 16) & 0xFFFFu) | (1u << 16);   // dim0 hi | tensor_dim1=1
    g1[3] = (nelem & 0xFFFFu) << 16;                  // tile_dim0@112
    g1[4] = 1u;                                       // tile_dim1 = 1
    g1[5] = nelem;                                    // tensor_dim0_stride lo
    g1[6] = 0u;
    g1[7] = 0u;
    asm volatile("tensor_load_to_lds %0, %1"
                 :: "s"(g0), "s"(g1)
                 : "memory");
}

// Async global->LDS copy of one 16B chunk per lane (GVS addressing),
// tracked on ASYNCcnt. saddr = uniform 64-bit base, voff = per-lane byte off.
__device__ __forceinline__ void async_copy_b128(unsigned lds_addr,
                                                unsigned voff,
                                                unsigned long long gbase)
{
    asm volatile("global_load_async_to_lds_b128 %0, %1, %2"
                 :: "v"(lds_addr), "v"(voff), "s"(gbase)
                 : "memory");
}

__device__ __forceinline__ void wait_asynccnt0()
{
    asm volatile("s_wait_asynccnt 0x0" ::: "memory");
}

// ---- wave-level 16x16 tile GEMM on LDS matrices (nmat = 64 or 32) ---------
// D = A * B  (row-major, leading dim ld). One wave per 16x16 tile of D.
// Fragment layouts per CDNA5 ISA 7.12.2:
//   A (16x4):  lanes 0-15 hold M=0..15; VGPR0/1 = K pair; lane half selects K+2
//   B (4x16):  N striped across lanes; VGPR0/1 = K rows, lane half selects K+2
//   C/D:       VGPR e -> M = 16*tm + e (+8 for upper lane half), N = lane&15
__device__ __forceinline__ void tile_mm(float* __restrict__ D,
                                        const float* __restrict__ A,
                                        const float* __restrict__ Bm,
                                        int nmat, int ld, int wave, int lane)
{
    const int tpd    = nmat >> 4;
    const int ntiles = tpd * tpd;
    if (wave < ntiles) {            // wave-uniform branch: EXEC stays all-ones
        const int tm = wave / tpd, tn = wave % tpd;
        const int lh = lane & 15, hh = lane >> 4;
        v8f acc = {};
        const int arow = (tm * 16 + lh) * ld;
        const int bcol = tn * 16 + lh;
        for (int k0 = 0; k0 < nmat; k0 += 4) {
            const int kb = k0 + (hh << 1);
            v2f av, bv;
            av.x = A[arow + kb];
            av.y = A[arow + kb + 1];
            bv.x = Bm[kb * ld + bcol];
            bv.y = Bm[(kb + 1) * ld + bcol];
            acc = __builtin_amdgcn_wmma_f32_16x16x4_f32(
                false, av, false, bv, (short)0, acc, false, false);
        }
        const int dr = tm * 16 + hh * 8;
        const int dc = tn * 16 + lh;
#pragma unroll
        for (int e = 0; e < 8; ++e)
            D[(dr + e) * ld + dc] = acc[e];
    }
}

// ---- kernel 1: SPD 5x5 kernels, transposed layout [c*25+p*5+q][r] ---------
__global__ void k_wker(const float* __restrict__ V, float* __restrict__ wkt)
{
    const int t = threadIdx.x;
    if (t < 128) {
        const int r = t >> 3, cch = t & 7;
        const float* vb = V + (r * 8 + cch) * 25;   // V[r][c][a][p]
        for (int p = 0; p < 5; ++p)
            for (int q = 0; q < 5; ++q) {
                float s = (p == q) ? EPS : 0.f;
                for (int a = 0; a < 5; ++a)
                    s = fmaf(vb[a * 5 + p], vb[a * 5 + q], s);
                wkt[(cch * 25 + p * 5 + q) * 16 + r] = s;
            }
    }
}

// ---- kernel 2: implicit-GEMM conv. M=3600 pixels, N=16 chans, K=200 -------
__global__ void __launch_bounds__(512)
k_conv(const float* __restrict__ X, const float* __restrict__ wkt,
       float* __restrict__ Yc)
{
    __shared__ __align__(16) float Blds[200 * 16];
    const int tid = threadIdx.x;
    const int b   = blockIdx.x;
    const int wave = tid >> 5, lane = tid & 31;

    // Stage B matrix (200x16 f32 = 12.8KB contiguous) with one TDM descriptor
    // issued by wave 0; wait on TENSORcnt, then workgroup barrier.
    if (wave == 0) {
        tdm_load_1d(lds_offset(Blds), wkt, 200 * 16);
        __builtin_amdgcn_s_wait_tensorcnt(0);
    }
    __syncthreads();

    const int lh = lane & 15, hh = lane >> 4;
    const float* Xb = X + (size_t)b * (8 * 64 * 64);

    for (int tile = wave; tile < 225; tile += 16) {
        const int m  = tile * 16 + lh;      // output pixel for A rows
        const int oi = m / 60, oj = m - oi * 60;
        v8f acc = {};
        for (int k0 = 0; k0 < 200; k0 += 4) {
            const int ka = k0 + (hh << 1);
            const int ka1 = ka + 1;
            const int c0 = ka / 25,  r0 = ka - c0 * 25;
            const int p0 = r0 / 5,   q0 = r0 - p0 * 5;
            const int c1 = ka1 / 25, r1 = ka1 - c1 * 25;
            const int p1 = r1 / 5,   q1 = r1 - p1 * 5;
            v2f av, bv;
            av.x = Xb[(c0 * 64 + oi + p0) * 64 + (oj + q0)];
            av.y = Xb[(c1 * 64 + oi + p1) * 64 + (oj + q1)];
            bv.x = Blds[ka  * 16 + lh];
            bv.y = Blds[ka1 * 16 + lh];
            acc = __builtin_amdgcn_wmma_f32_16x16x4_f32(
                false, av, false, bv, (short)0, acc, false, false);
        }
        const int mr = tile * 16 + hh * 8;  // D rows (pixels), D col = channel
#pragma unroll
        for (int e = 0; e < 8; ++e)
            Yc[((size_t)b * 16 + lh) * 3600 + (mr + e)] = acc[e];
    }
}

// ---- kernel 3: per-channel 1/||out||_F, fixed-order reduction -------------
__global__ void k_norm(const float* __restrict__ Yc, float* __restrict__ invn)
{
    __shared__ float red[256];
    const int r = blockIdx.x, tid = threadIdx.x;
    float s = 0.f;
    for (int b = 0; b < 512; ++b) {
        const float* p = Yc + ((size_t)b * 16 + r) * 3600;
        for (int pix = tid; pix < 3600; pix += 256) {
            const float v = p[pix];
            s = fmaf(v, v, s);
        }
    }
    red[tid] = s;
    __syncthreads();
    for (int off = 128; off > 0; off >>= 1) {
        if (tid < off) red[tid] += red[tid + off];
        __syncthreads();
    }
    if (tid == 0) invn[r] = rsqrtf(red[0] + 1e-30f);
}

// ---- kernel 4: per-(b,r) matrix log -> pool -> matrix exp -> tril features
__global__ void __launch_bounds__(512)
k_matfunc(const float* __restrict__ Yc, const float* __restrict__ invn,
          float* __restrict__ vec)
{
    __shared__ __align__(16) float SB0[64 * LDA];
    __shared__ __align__(16) float SB1[64 * LDA];
    __shared__ __align__(16) float SB2[64 * LDA];
    __shared__ __align__(16) float SB3[64 * LDA];
    __shared__ float s_scal;

    const int tid  = threadIdx.x;
    const int wave = tid >> 5, lane = tid & 31;
    const int bb = blockIdx.x >> 4;    // batch
    const int rr = blockIdx.x & 15;    // out channel
    const float inv = invn[rr];
    const float* Y = Yc + ((size_t)bb * 16 + rr) * 3600;

    // Stage raw 60x60 tile (3600 f32 = 900 x 16B, 16B-aligned) into SB3 via
    // async copy: one coalesced HBM sweep; symmetrize reads then hit LDS.
    {
        const unsigned lbase = lds_offset(SB3);
        const unsigned long long gbase = (unsigned long long)Y;
        for (int i = tid; i < 900; i += 512)
            async_copy_b128(lbase + i * 16u, (unsigned)(i * 16), gbase);
        wait_asynccnt0();
    }
    __syncthreads();

    // S = sym(relu(out/norm)) + eps*I, identity-padded 60 -> 64
    for (int idx = tid; idx < 64 * 64; idx += 512) {
        const int i = idx >> 6, j = idx & 63;
        float v;
        if (i < 60 && j < 60) {
            float a = SB3[i * 60 + j]; a = a > 0.f ? a : 0.f;
            float b = SB3[j * 60 + i]; b = b > 0.f ? b : 0.f;
            v = 0.5f * (a + b) * inv + ((i == j) ? EPS : 0.f);
        } else {
            v = (i == j) ? 1.f : 0.f;
        }
        SB0[i * LDA + j] = v;
    }
    __syncthreads();

    // ---- log(S) via inverse scaling & squaring (N_SQRT coupled NS sqrts) --
    for (int sq = 0; sq < N_SQRT; ++sq) {
        if (tid == 0) {
            float tr = 0.f;
            for (int i = 0; i < 64; ++i) tr += SB0[i * LDA + i];
            s_scal = tr + 1e-6f;            // trace >= lambda_max for SPD
        }
        __syncthreads();
        const float cs  = s_scal;
        const float icv = 1.f / cs;
        // Yn = A/c (in place in SB0), Z = I (SB1)
        for (int idx = tid; idx < 64 * 64; idx += 512) {
            const int i = idx >> 6, j = idx & 63;
            SB0[i * LDA + j] *= icv;
            SB1[i * LDA + j] = (i == j) ? 1.f : 0.f;
        }
        __syncthreads();
        for (int it = 0; it < NS_ITERS; ++it) {
            tile_mm(SB2, SB1, SB0, 64, LDA, wave, lane);   // T = Z*Yn
            __syncthreads();
            tile_mm(SB3, SB0, SB2, 64, LDA, wave, lane);   // tmp = Yn*T
            __syncthreads();
            for (int idx = tid; idx < 64 * 64; idx += 512) {
                const int o = (idx >> 6) * LDA + (idx & 63);
                SB0[o] = 1.5f * SB0[o] - 0.5f * SB3[o];    // Yn <- (3Yn - Yn T)/2
            }
            __syncthreads();
            tile_mm(SB3, SB2, SB1, 64, LDA, wave, lane);   // tmp = T*Z
            __syncthreads();
            for (int idx = tid; idx < 64 * 64; idx += 512) {
                const int o = (idx >> 6) * LDA + (idx & 63);
                SB1[o] = 1.5f * SB1[o] - 0.5f * SB3[o];    // Z <- (3Z - T Z)/2
            }
            __syncthreads();
        }
        const float rs = sqrtf(cs);
        for (int idx = tid; idx < 64 * 64; idx += 512) {
            const int o = (idx >> 6) * LDA + (idx & 63);
            SB0[o] *= rs;                                  // A <- sqrt(A)
        }
        __syncthreads();
    }

    // E = A_s - I ; log(A_s) ~= E - E^2/2 + E^3/3 - E^4/4  (Mercator)
    for (int idx = tid; idx < 64 * 64; idx += 512) {
        const int i = idx >> 6, j = idx & 63;
        if (i == j) SB0[i * LDA + j] -= 1.f;
    }
    __syncthreads();
    tile_mm(SB2, SB0, SB0, 64, LDA, wave, lane);           // E^2
    __syncthreads();
    for (int idx = tid; idx < 64 * 64; idx += 512) {
        const int o = (idx >> 6) * LDA + (idx & 63);
        SB1[o] = SB0[o] - 0.5f * SB2[o];
    }
    __syncthreads();
    tile_mm(SB3, SB2, SB0, 64, LDA, wave, lane);           // E^3
    __syncthreads();
    for (int idx = tid; idx < 64 * 64; idx += 512) {
        const int o = (idx >> 6) * LDA + (idx & 63);
        SB1[o] += (1.f / 3.f) * SB3[o];
    }
    __syncthreads();
    tile_mm(SB0, SB2, SB2, 64, LDA, wave, lane);           // E^4 (E dead)
    __syncthreads();
    for (int idx = tid; idx < 64 * 64; idx += 512) {
        const int o = (idx >> 6) * LDA + (idx & 63);
        SB1[o] -= 0.25f * SB0[o];                          // acc = log(A_s)
    }
    __syncthreads();

    // ---- pool (2x2 avg over top-left 60x60) * 2^N_SQRT, /2^EXP_T, pad 32 --
    for (int idx = tid; idx < 32 * 32; idx += 512)
        SB0[(idx >> 5) * LDB + (idx & 31)] = 0.f;
    __syncthreads();
    const float pscale = (float)(1 << N_SQRT) * 0.25f / (float)(1 << EXP_T);
    for (int idx = tid; idx < 30 * 30; idx += 512) {
        const int i = idx / 30, j = idx - i * 30;
        const float s = SB1[(2 * i) * LDA + 2 * j]     + SB1[(2 * i) * LDA + 2 * j + 1]
                      + SB1[(2 * i + 1) * LDA + 2 * j] + SB1[(2 * i + 1) * LDA + 2 * j + 1];
        SB0[i * LDB + j] = pscale * s;                     // U = pooled / 2^t
    }
    __syncthreads();

    // ---- exp(P) = (I + U + U^2/2 + U^3/6 + U^4/24)^(2^EXP_T) --------------
    tile_mm(SB2, SB0, SB0, 32, LDB, wave, lane);           // U^2
    __syncthreads();
    tile_mm(SB3, SB2, SB0, 32, LDB, wave, lane);           // U^3
    __syncthreads();
    tile_mm(SB1, SB2, SB2, 32, LDB, wave, lane);           // U^4
    __syncthreads();
    for (int idx = tid; idx < 32 * 32; idx += 512) {
        const int i = idx >> 5, j = idx & 31;
        const int o = i * LDB + j;
        SB0[o] = ((i == j) ? 1.f : 0.f) + SB0[o] + 0.5f * SB2[o]
               + (1.f / 6.f) * SB3[o] + (1.f / 24.f) * SB1[o];
    }
    __syncthreads();
    float* cur = SB0;
    float* oth = SB1;
    for (int t = 0; t < EXP_T; ++t) {
        tile_mm(oth, cur, cur, 32, LDB, wave, lane);       // E <- E*E
        __syncthreads();
        float* tmp = cur; cur = oth; oth = tmp;
    }

    // ---- tril(30) features, row-major pairs, channel-major concat ---------
    for (int t = tid; t < 465; t += 512) {
        int i = (int)((sqrtf(8.f * (float)t + 1.f) - 1.f) * 0.5f);
        while ((i + 1) * (i + 2) / 2 <= t) ++i;
        while (i * (i + 1) / 2 > t) --i;
        const int j = t - i * (i + 1) / 2;
        vec[(size_t)bb * 7440 + rr * 465 + t] = cur[i * LDB + j];
    }
}

// ---- kernel 5: classifier out[b,cls] = b_cls + vec[b,:] . W_cls[cls,:] ----
__global__ void k_cls(const float* __restrict__ vec, const float* __restrict__ Wc,
                      const float* __restrict__ bc, float* __restrict__ out)
{
    const int idx = blockIdx.x * 256 + threadIdx.x;
    if (idx < 5120) {
        const int b = idx / 10, cls = idx - b * 10;
        const float* v = vec + (size_t)b * 7440;
        const float* w = Wc + (size_t)cls * 7440;
        float a = bc[cls];
        for (int f0 = 0; f0 < 7440; f0 += 240) {
            if (f0 + 240 < 7440) {
                __builtin_prefetch(v + f0 + 240, 0, 1);   // global_prefetch_b8
                __builtin_prefetch(w + f0 + 240, 0, 1);
            }
            for (int f = f0; f < f0 + 240; ++f)
                a = fmaf(v[f], w[f], a);
        }
        out[idx] = a;
    }
}

extern "C" void kernel_launch(void* const* d_in, const int* in_sizes, int n_in,
                              void* d_out, int out_size, void* d_ws, size_t ws_size,
                              hipStream_t stream)
{
    (void)in_sizes; (void)n_in; (void)out_size; (void)ws_size;
    const float* X  = (const float*)d_in[0];   // [512,8,64,64]
    const float* V  = (const float*)d_in[1];   // [16,8,5,5]
    const float* Wc = (const float*)d_in[2];   // [10,7440]
    const float* bc = (const float*)d_in[3];   // [10]
    float* ws   = (float*)d_ws;
    float* wkt  = ws;                          // 200*16
    float* Yc   = ws + 4096;                   // 512*16*3600 = 29,491,200
    float* invn = Yc + 29491200;               // 16
    float* vec  = invn + 64;                   // 512*7440 = 3,809,280
    float* out  = (float*)d_out;               // [512,10]

    k_wker   <<<dim3(1),    dim3(128), 0, stream>>>(V, wkt);
    k_conv   <<<dim3(512),  dim3(512), 0, stream>>>(X, wkt, Yc);
    k_norm   <<<dim3(16),   dim3(256), 0, stream>>>(Yc, invn);
    k_matfunc<<<dim3(8192), dim3(512), 0, stream>>>(Yc, invn, vec);
    k_cls    <<<dim3(20),   dim3(256), 0, stream>>>(vec, Wc, bc, out);
}